// GCNClassifier_32109175505557
// MI455X (gfx1250) — compile-verified
//
#include <hip/hip_runtime.h>
#include <stdint.h>

#define HID 128

typedef __attribute__((ext_vector_type(2))) float v2f;
typedef __attribute__((ext_vector_type(8))) float v8f;

// ---------------- utility ----------------
__global__ void zero_f32(float* __restrict__ p, long n) {
    long i = (long)blockIdx.x * blockDim.x + threadIdx.x;
    long stride = (long)gridDim.x * blockDim.x;
    for (; i < n; i += stride) p[i] = 0.0f;
}

// ---------------- degrees / norms ----------------
__global__ void deg_kernel(const int* __restrict__ src, const int* __restrict__ dst,
                           float* __restrict__ in_deg, float* __restrict__ out_deg, int E) {
    int e = blockIdx.x * blockDim.x + threadIdx.x;
    if (e < E) {
        atomicAdd(&in_deg[dst[e]], 1.0f);
        atomicAdd(&out_deg[src[e]], 1.0f);
    }
}

__global__ void norm_kernel(const float* __restrict__ in_deg, const float* __restrict__ out_deg,
                            float* __restrict__ norm_src, float* __restrict__ norm_dst, int N) {
    int n = blockIdx.x * blockDim.x + threadIdx.x;
    if (n < N) {
        norm_src[n] = rsqrtf(fmaxf(out_deg[n], 1.0f));
        norm_dst[n] = rsqrtf(fmaxf(in_deg[n], 1.0f));
    }
}

// ---------------- layer 0 (HID_in = 1) ----------------
__global__ void scatter0(const float* __restrict__ in_deg, const float* __restrict__ norm_src,
                         const int* __restrict__ src, const int* __restrict__ dst,
                         float* __restrict__ agg0, int E) {
    int e = blockIdx.x * blockDim.x + threadIdx.x;
    if (e < E) {
        int s = src[e];
        atomicAdd(&agg0[dst[e]], in_deg[s] * norm_src[s]);
    }
}

// out[n][f] = relu(agg0[n]*norm_dst[n]*W0[f] + b0[f]) * norm_src[n]   (pre-scaled for next layer)
__global__ void layer0_kernel(const float* __restrict__ agg0, const float* __restrict__ norm_dst,
                              const float* __restrict__ norm_src, const float* __restrict__ W0,
                              const float* __restrict__ b0, float* __restrict__ out, int N) {
    int i = blockIdx.x * blockDim.x + threadIdx.x;
    if (i < N * HID) {
        int n = i >> 7, f = i & (HID - 1);
        float v = fmaxf(agg0[n] * norm_dst[n] * W0[f] + b0[f], 0.0f);
        out[i] = v * norm_src[n];
    }
}

// ---------------- 128-wide edge scatter: agg[dst] += hs[src] ----------------
// one wave per edge; lane handles 4 consecutive features (float4 gather, 4 f32 atomics)
__global__ void scatter128(const float* __restrict__ hs, const int* __restrict__ src,
                           const int* __restrict__ dst, float* __restrict__ agg, int E) {
    int e = blockIdx.x * (blockDim.x >> 5) + (threadIdx.x >> 5);
    if (e >= E) return;
    int lane = threadIdx.x & 31;
    int s = src[e], d = dst[e];
    const float4 v = ((const float4*)(hs + (long)s * HID))[lane];
    float* o = agg + (long)d * HID + lane * 4;
    atomicAdd(o + 0, v.x);
    atomicAdd(o + 1, v.y);
    atomicAdd(o + 2, v.z);
    atomicAdd(o + 3, v.w);
}

// ---------------- WMMA GEMM: out = act((A .* norm_dst) @ W + b) [* norm_src] ----------------
// A: N x 128, W: 128 x 128. Block = 256 threads = 8 waves; block -> 16-row tile,
// wave w -> 16-col tile. K-loop: 32x V_WMMA_F32_16X16X4_F32.
__global__ void gcn_gemm_wmma(const float* __restrict__ A, const float* __restrict__ W,
                              const float* __restrict__ b, const float* __restrict__ norm_dst,
                              const float* __restrict__ norm_src, float* __restrict__ out,
                              int N, int apply_src_scale) {
    const int lane = threadIdx.x & 31;
    const int wave = threadIdx.x >> 5;          // column tile 0..7
    const int rowBase = blockIdx.x * 16;
    const int colBase = wave * 16;
    const int m  = lane & 15;                   // A row / D col within tile
    const int kg = lane >> 4;                   // lane half selects K pair / D row half

    int arow = rowBase + m; if (arow >= N) arow = N - 1;   // clamped loads
    const float nd = norm_dst[arow];
    const float* Arow = A + (long)arow * HID;

    v8f acc = {0.f, 0.f, 0.f, 0.f, 0.f, 0.f, 0.f, 0.f};
#pragma unroll
    for (int k0 = 0; k0 < HID; k0 += 4) {
        const int k = k0 + 2 * kg;              // per ISA: VGPR v holds K = 2*(lane/16)+v
        v2f a, bm;
        a.x  = Arow[k]     * nd;
        a.y  = Arow[k + 1] * nd;
        bm.x = W[(long)k * HID + colBase + m];
        bm.y = W[(long)(k + 1) * HID + colBase + m];
        acc = __builtin_amdgcn_wmma_f32_16x16x4_f32(false, a, false, bm,
                                                    (short)0, acc, false, false);
    }

    const int col = colBase + m;                // D: lane -> N = lane%16
    const float bias = b[col];
#pragma unroll
    for (int r = 0; r < 8; ++r) {               // D: VGPR r -> M = r + 8*(lane/16)
        int mrow = rowBase + r + 8 * kg;
        if (mrow < N) {
            float v = fmaxf(acc[r] + bias, 0.0f);
            if (apply_src_scale) v *= norm_src[mrow];
            out[(long)mrow * HID + col] = v;
        }
    }
}

// ---------------- graph pooling ----------------
__global__ void pool_kernel(const float* __restrict__ h, const int* __restrict__ gid,
                            float* __restrict__ hg_sum, float* __restrict__ cnt, int N) {
    int node = blockIdx.x * (blockDim.x >> 5) + (threadIdx.x >> 5);
    if (node >= N) return;
    int lane = threadIdx.x & 31;
    int g = gid[node];
    const float4 v = ((const float4*)(h + (long)node * HID))[lane];
    float* o = hg_sum + (long)g * HID + lane * 4;
    atomicAdd(o + 0, v.x);
    atomicAdd(o + 1, v.y);
    atomicAdd(o + 2, v.z);
    atomicAdd(o + 3, v.w);
    if (lane == 0) atomicAdd(&cnt[g], 1.0f);
}

// ---------------- classifier: out = (hg_sum/cnt) @ Wc + bc,  Wc: 128 x C ----------------
// one wave per 16-graph tile; column mask handles C=10 < 16.
__global__ void classifier_wmma(const float* __restrict__ hg_sum, const float* __restrict__ cnt,
                                const float* __restrict__ Wc, const float* __restrict__ bc,
                                float* __restrict__ out, int G, int C) {
    const int lane = threadIdx.x & 31;
    const int rowBase = blockIdx.x * 16;
    const int m  = lane & 15;
    const int kg = lane >> 4;

    int grow = rowBase + m; if (grow >= G) grow = G - 1;
    const float inv = 1.0f / fmaxf(cnt[grow], 1.0f);
    const float* Arow = hg_sum + (long)grow * HID;

    v8f acc = {0.f, 0.f, 0.f, 0.f, 0.f, 0.f, 0.f, 0.f};
#pragma unroll
    for (int k0 = 0; k0 < HID; k0 += 4) {
        const int k = k0 + 2 * kg;
        v2f a, bm;
        a.x  = Arow[k]     * inv;
        a.y  = Arow[k + 1] * inv;
        bm.x = (m < C) ? Wc[(long)k * C + m] : 0.0f;
        bm.y = (m < C) ? Wc[(long)(k + 1) * C + m] : 0.0f;
        acc = __builtin_amdgcn_wmma_f32_16x16x4_f32(false, a, false, bm,
                                                    (short)0, acc, false, false);
    }
    const float bias = (m < C) ? bc[m] : 0.0f;
#pragma unroll
    for (int r = 0; r < 8; ++r) {
        int g = rowBase + r + 8 * kg;
        if (g < G && m < C) out[(long)g * C + m] = acc[r] + bias;
    }
}

// ---------------- launch ----------------
extern "C" void kernel_launch(void* const* d_in, const int* in_sizes, int n_in,
                              void* d_out, int out_size, void* d_ws, size_t ws_size,
                              hipStream_t stream) {
    const float* W0 = (const float*)d_in[0];
    const float* b0 = (const float*)d_in[1];
    const float* W1 = (const float*)d_in[2];
    const float* b1 = (const float*)d_in[3];
    const float* W2 = (const float*)d_in[4];
    const float* b2 = (const float*)d_in[5];
    const float* Wc = (const float*)d_in[6];
    const float* bc = (const float*)d_in[7];
    const int* src  = (const int*)d_in[8];
    const int* dst  = (const int*)d_in[9];
    const int* gid  = (const int*)d_in[10];

    const int E = in_sizes[8];
    const int N = in_sizes[10];
    const int C = in_sizes[7];          // 10
    const int G = out_size / C;         // 512

    // workspace carve-up (all offsets stay 16B-aligned for the float4 paths)
    float* ws = (float*)d_ws;
    float* in_deg  = ws;  ws += N;
    float* out_deg = ws;  ws += N;
    float* nsrc    = ws;  ws += N;
    float* ndst    = ws;  ws += N;
    float* agg0    = ws;  ws += N;
    ws = (float*)(((uintptr_t)ws + 15) & ~(uintptr_t)15);
    float* Hbuf    = ws;  ws += (long)N * HID;   // pre-scaled node features
    float* Abuf    = ws;  ws += (long)N * HID;   // aggregation accumulator
    float* hg_sum  = ws;  ws += (long)G * HID;
    float* cnt     = ws;  ws += G;

    float* out = (float*)d_out;

    const int T = 256;
    const int rowTiles = (N + 15) / 16;
    const int edgeWaveBlocks = (E + 7) / 8;      // 8 waves / block, wave per edge
    const int nodeWaveBlocks = (N + 7) / 8;

    // degrees + norms
    zero_f32<<<2048, T, 0, stream>>>(in_deg, (long)5 * N);   // in/out_deg, norms, agg0
    deg_kernel<<<(E + T - 1) / T, T, 0, stream>>>(src, dst, in_deg, out_deg, E);
    norm_kernel<<<(N + T - 1) / T, T, 0, stream>>>(in_deg, out_deg, nsrc, ndst, N);

    // layer 0 (input feature = in_deg, HID_in = 1)
    scatter0<<<(E + T - 1) / T, T, 0, stream>>>(in_deg, nsrc, src, dst, agg0, E);
    layer0_kernel<<<((long)N * HID + T - 1) / T, T, 0, stream>>>(agg0, ndst, nsrc, W0, b0, Hbuf, N);

    // layer 1
    zero_f32<<<4096, T, 0, stream>>>(Abuf, (long)N * HID);
    scatter128<<<edgeWaveBlocks, T, 0, stream>>>(Hbuf, src, dst, Abuf, E);
    gcn_gemm_wmma<<<rowTiles, T, 0, stream>>>(Abuf, W1, b1, ndst, nsrc, Hbuf, N, 1);

    // layer 2 (no src pre-scale: output feeds pooling)
    zero_f32<<<4096, T, 0, stream>>>(Abuf, (long)N * HID);
    scatter128<<<edgeWaveBlocks, T, 0, stream>>>(Hbuf, src, dst, Abuf, E);
    gcn_gemm_wmma<<<rowTiles, T, 0, stream>>>(Abuf, W2, b2, ndst, nsrc, Hbuf, N, 0);

    // mean pooling + classifier
    zero_f32<<<512, T, 0, stream>>>(hg_sum, (long)G * HID + G);  // hg_sum and cnt contiguous
    pool_kernel<<<nodeWaveBlocks, T, 0, stream>>>(Hbuf, gid, hg_sum, cnt, N);
    classifier_wmma<<<(G + 15) / 16, 32, 0, stream>>>(hg_sum, cnt, Wc, bc, out, G, C);
}